// WeightsAsInputsExpert_43104291782790
// MI455X (gfx1250) — compile-verified
//
#include <hip/hip_runtime.h>
#include <math.h>

#define HIDDEN 2048
#define INTER  8192
#define TOPK   8
#define ICHUNKS 16

typedef float v2f __attribute__((ext_vector_type(2)));
typedef float v4f __attribute__((ext_vector_type(4)));
typedef float v8f __attribute__((ext_vector_type(8)));

static __device__ __forceinline__ v2f lo2(v4f v) { return __builtin_shufflevector(v, v, 0, 1); }
static __device__ __forceinline__ v2f hi2(v4f v) { return __builtin_shufflevector(v, v, 2, 3); }

// ---------------------------------------------------------------------------
// Kernel 1: gate/up GEMV via V_WMMA_F32_16X16X4_F32 + SwiGLU, pre-scaled by
// weights_topk[k].  One wave per 16-row tile of INTER for one expert.
// b128 loads: lanes 0-15 fetch W[row, h0..h0+3], lanes 16-31 fetch
// W[row, h0+4..h0+7]; the two WMMAs per chunk use K-slot assignments
// {h0,h0+1,h0+4,h0+5} and {h0+2,h0+3,h0+6,h0+7}, with B built from the same
// float4 of x so A and B K-slots agree.
// ---------------------------------------------------------------------------
__global__ __launch_bounds__(256) void k1_gate_up_swiglu(
    const float* __restrict__ x,        // (HIDDEN)
    const float* __restrict__ gate_w,   // (TOPK, INTER, HIDDEN)
    const float* __restrict__ up_w,     // (TOPK, INTER, HIDDEN)
    const float* __restrict__ wtopk,    // (TOPK)
    float* __restrict__ inter_scaled)   // (TOPK, INTER)
{
    const int lane = threadIdx.x & 31;
    const int wave = threadIdx.x >> 5;
    const int tile = blockIdx.x * (blockDim.x >> 5) + wave;   // 0..4095

    const int tiles_per_expert = INTER / 16;                  // 512
    const int k  = tile / tiles_per_expert;
    const int i0 = (tile - k * tiles_per_expert) * 16;

    const int g   = lane >> 4;    // lane-half: selects h0..h0+3 vs h0+4..h0+7
    const int row = lane & 15;    // A-matrix row

    const size_t wbase = (size_t)k * INTER * HIDDEN + (size_t)(i0 + row) * HIDDEN + 4 * g;
    const float* __restrict__ Ag = gate_w + wbase;
    const float* __restrict__ Au = up_w   + wbase;
    const float* __restrict__ Bx = x + 4 * g;

    v8f acc_g = {0.f, 0.f, 0.f, 0.f, 0.f, 0.f, 0.f, 0.f};
    v8f acc_u = {0.f, 0.f, 0.f, 0.f, 0.f, 0.f, 0.f, 0.f};

#pragma unroll 2
    for (int h0 = 0; h0 < HIDDEN; h0 += 8) {
        v4f ag = __builtin_nontemporal_load((const v4f*)(Ag + h0));
        v4f au = __builtin_nontemporal_load((const v4f*)(Au + h0));
        v4f b  = *(const v4f*)(Bx + h0);
        acc_g = __builtin_amdgcn_wmma_f32_16x16x4_f32(
            false, lo2(ag), false, lo2(b), (short)0, acc_g, false, false);
        acc_u = __builtin_amdgcn_wmma_f32_16x16x4_f32(
            false, lo2(au), false, lo2(b), (short)0, acc_u, false, false);
        acc_g = __builtin_amdgcn_wmma_f32_16x16x4_f32(
            false, hi2(ag), false, hi2(b), (short)0, acc_g, false, false);
        acc_u = __builtin_amdgcn_wmma_f32_16x16x4_f32(
            false, hi2(au), false, hi2(b), (short)0, acc_u, false, false);
    }

    // D: VGPR r holds row (r + 8*g); all 16 columns identical -> lanes with
    // (lane&15)==0 (i.e. lanes 0 and 16) carry the 16 results.
    if (row == 0) {
        const float wk = wtopk[k];
#pragma unroll
        for (int r = 0; r < 8; ++r) {
            float gv = acc_g[r];
            float uv = acc_u[r];
            float s  = gv / (1.0f + __expf(-gv));     // SiLU
            inter_scaled[(size_t)k * INTER + i0 + 8 * g + r] = wk * s * uv;
        }
    }
}

// ---------------------------------------------------------------------------
// Kernel 2: down projection partials.  Grid covers (expert k, 16-row h-tile,
// i-chunk); topk weight already folded into inter_scaled.  Same b128 WMMA-GEMV
// scheme.  Deterministic: each wave writes its own 16 partials.
// ---------------------------------------------------------------------------
__global__ __launch_bounds__(256) void k2_down_partial(
    const float* __restrict__ down_w,       // (TOPK, HIDDEN, INTER)
    const float* __restrict__ inter_scaled, // (TOPK, INTER)
    float* __restrict__ partials)           // (TOPK*ICHUNKS, HIDDEN)
{
    const int lane = threadIdx.x & 31;
    const int wave = threadIdx.x >> 5;
    const int tile = blockIdx.x * (blockDim.x >> 5) + wave;   // 0..16383

    const int H_TILES = HIDDEN / 16;                          // 128
    const int ic = tile % ICHUNKS;
    const int t2 = tile / ICHUNKS;
    const int ht = t2 % H_TILES;
    const int k  = t2 / H_TILES;
    const int h0 = ht * 16;
    const int iStart = ic * (INTER / ICHUNKS);                // 512-wide chunks

    const int g   = lane >> 4;
    const int row = lane & 15;

    const float* __restrict__ A = down_w + ((size_t)k * HIDDEN + h0 + row) * INTER + 4 * g;
    const float* __restrict__ B = inter_scaled + (size_t)k * INTER + 4 * g;

    v8f acc = {0.f, 0.f, 0.f, 0.f, 0.f, 0.f, 0.f, 0.f};

#pragma unroll 4
    for (int i = iStart; i < iStart + INTER / ICHUNKS; i += 8) {
        v4f a = __builtin_nontemporal_load((const v4f*)(A + i));
        v4f b = *(const v4f*)(B + i);
        acc = __builtin_amdgcn_wmma_f32_16x16x4_f32(
            false, lo2(a), false, lo2(b), (short)0, acc, false, false);
        acc = __builtin_amdgcn_wmma_f32_16x16x4_f32(
            false, hi2(a), false, hi2(b), (short)0, acc, false, false);
    }

    if (row == 0) {
        float* dst = partials + (size_t)(k * ICHUNKS + ic) * HIDDEN + h0 + 8 * g;
#pragma unroll
        for (int r = 0; r < 8; ++r) dst[r] = acc[r];
    }
}

// ---------------------------------------------------------------------------
// Kernel 3: reduce 128 partial vectors into the output (writes every element,
// so no prior memset is required).
// ---------------------------------------------------------------------------
__global__ __launch_bounds__(256) void k3_reduce(
    const float* __restrict__ partials,  // (TOPK*ICHUNKS, HIDDEN)
    float* __restrict__ out)             // (HIDDEN)
{
    const int h = blockIdx.x * blockDim.x + threadIdx.x;
    if (h < HIDDEN) {
        float s = 0.f;
#pragma unroll 8
        for (int j = 0; j < TOPK * ICHUNKS; ++j)
            s += partials[(size_t)j * HIDDEN + h];
        out[h] = s;
    }
}

// ---------------------------------------------------------------------------
extern "C" void kernel_launch(void* const* d_in, const int* in_sizes, int n_in,
                              void* d_out, int out_size, void* d_ws, size_t ws_size,
                              hipStream_t stream) {
    const float* x  = (const float*)d_in[0];   // (1, HIDDEN, 1, 1)
    const float* gw = (const float*)d_in[1];   // (TOPK, INTER, HIDDEN)
    const float* uw = (const float*)d_in[2];   // (TOPK, INTER, HIDDEN)
    const float* dw = (const float*)d_in[3];   // (TOPK, HIDDEN, INTER)
    const float* wt = (const float*)d_in[4];   // (TOPK,)
    float* out = (float*)d_out;

    float* inter    = (float*)d_ws;                 // TOPK*INTER floats (256 KB)
    float* partials = inter + (size_t)TOPK * INTER; // TOPK*ICHUNKS*HIDDEN floats (1 MB)

    // k1: 8 experts * 512 tiles = 4096 waves, 8 waves/block
    k1_gate_up_swiglu<<<512, 256, 0, stream>>>(x, gw, uw, wt, inter);

    // k2: 8 * 128 * 16 = 16384 waves, 8 waves/block
    k2_down_partial<<<2048, 256, 0, stream>>>(dw, inter, partials);

    // k3: 2048 outputs
    k3_reduce<<<HIDDEN / 256, 256, 0, stream>>>(partials, out);
}